// ODEBlock_72335839199611
// MI455X (gfx1250) — compile-verified
//
#include <hip/hip_runtime.h>
#include <hip/hip_bf16.h>

typedef __attribute__((ext_vector_type(2))) float v2f;
typedef __attribute__((ext_vector_type(8))) float v8f;

#define H1      64
#define MROWS   4000
#define TPTS    100
#define CSTRIDE 67
#define GAMMA_C 0.1f

// Head projection: D(16x16) = Z(16x64) x W3^T(64x16, cols 4..15 zero), via 16
// chained V_WMMA_F32_16X16X4_F32 (two independent chains for ILP). Then per-lane
// relu/scale and a 4-wide xor-shuffle reduction over the head units.
__device__ __forceinline__ void head_wmma(const float (&Z)[32], const float (&WB)[32],
                                          float b3n, float ws2n, float bs2v,
                                          float (&outv)[8]) {
  v8f acc0 = {0.f, 0.f, 0.f, 0.f, 0.f, 0.f, 0.f, 0.f};
  v8f acc1 = {0.f, 0.f, 0.f, 0.f, 0.f, 0.f, 0.f, 0.f};
#pragma unroll
  for (int c = 0; c < 16; c += 2) {
    v2f a0, b0, a1, b1;
    a0.x = Z[2 * c];      a0.y = Z[2 * c + 1];
    b0.x = WB[2 * c];     b0.y = WB[2 * c + 1];
    a1.x = Z[2 * c + 2];  a1.y = Z[2 * c + 3];
    b1.x = WB[2 * c + 2]; b1.y = WB[2 * c + 3];
    acc0 = __builtin_amdgcn_wmma_f32_16x16x4_f32(false, a0, false, b0,
                                                 (short)0, acc0, false, false);
    acc1 = __builtin_amdgcn_wmma_f32_16x16x4_f32(false, a1, false, b1,
                                                 (short)0, acc1, false, false);
  }
#pragma unroll
  for (int v = 0; v < 8; ++v) {
    float d = acc0[v] + acc1[v];
    float h = fmaxf(d + b3n, 0.f) * ws2n;   // zero for lanes with n >= 4
    h += __shfl_xor(h, 1, 32);
    h += __shfl_xor(h, 2, 32);              // lanes {0..3},{16..19} now hold sum
    outv[v] = h + bs2v;
  }
}

__global__ __launch_bounds__(64) void sir_node_fused_kernel(
    const float* __restrict__ x,   const float* __restrict__ t,
    const float* __restrict__ Ws1, const float* __restrict__ bs1,
    const float* __restrict__ W3,  const float* __restrict__ b3,
    const float* __restrict__ Ws2, const float* __restrict__ bs2,
    float* __restrict__ out)
{
  const int lane  = threadIdx.x & 31;
  const int wave  = blockIdx.x * 2 + (threadIdx.x >> 5);
  const int mbase = wave * 16;                 // grid sized exactly: mbase < 4000

  // WMMA A-layout mapping: lane L holds rows m = L%16; slot (2c+j) holds
  // K = 4c + 2*(L>=16) + j.
  const int mrow = lane & 15;
  const int hi   = (lane >> 4) * 2;
  const int m    = mbase + mrow;
  const int n    = lane & 15;                  // B/C/D column index

  float S[32], I[32], R[32], BETA[32], WB[32];

  const float S0v = x[m * CSTRIDE + 0];
  const float I0v = x[m * CSTRIDE + 1];

#pragma unroll
  for (int rr = 0; rr < 32; ++rr) {
    const int h = 4 * (rr >> 1) + hi + (rr & 1);
    const float w = Ws1[h];
    const float b = bs1[h];
    S[rr] = fmaxf(S0v * w + b, 0.f);
    I[rr] = fmaxf(I0v * w + b, 0.f);
    R[rr] = 0.f;
    const float bg = x[m * CSTRIDE + 3 + h];
    BETA[rr] = 0.5f / (1.f + __expf(-bg));     // 0.5 * sigmoid(bg), constant in t
    WB[rr] = (n < 4) ? W3[n * H1 + h] : 0.f;   // W3^T in WMMA B layout, zero-padded
  }

  const float b3n  = (n < 4) ? b3[n]  : 0.f;
  const float ws2n = (n < 4) ? Ws2[n] : 0.f;
  const float bs2v = bs2[0];

  for (int ts = 0; ts < TPTS; ++ts) {
    float aS[8], aI[8], aR[8];
    head_wmma(S, WB, b3n, ws2n, bs2v, aS);
    head_wmma(I, WB, b3n, ws2n, bs2v, aI);
    head_wmma(R, WB, b3n, ws2n, bs2v, aR);

    if ((lane & 15) == 0) {                    // writer lanes 0 and 16
      const int mloc = (lane >> 4) * 8;        // D layout: M = v + 8*(L/16)
#pragma unroll
      for (int v = 0; v < 8; ++v) {
        const int mg = mbase + mloc + v;
        const float sv = aS[v], iv = aI[v], rv = aR[v];
        const float mx = fmaxf(sv, fmaxf(iv, rv));
        const float es = __expf(sv - mx);
        const float ei = __expf(iv - mx);
        const float er = __expf(rv - mx);
        const float inv = 1.f / (es + ei + er);
        out[0 * TPTS * MROWS + ts * MROWS + mg] = es * inv;
        out[1 * TPTS * MROWS + ts * MROWS + mg] = ei * inv;
        out[2 * TPTS * MROWS + ts * MROWS + mg] = er * inv;
      }
    }

    if (ts < TPTS - 1) {
      const float dt  = t[ts + 1] - t[ts];
      const float hdt = 0.5f * dt;
      const float c6  = dt * (1.f / 6.f);
#pragma unroll
      for (int rr = 0; rr < 32; ++rr) {
        const float sv = S[rr], iv = I[rr], be = BETA[rr];
        const float inf1 = be * sv * iv;
        const float k1s = -inf1, k1i = inf1 - GAMMA_C * iv, k1r = GAMMA_C * iv;
        const float s2 = sv + hdt * k1s, i2 = iv + hdt * k1i;
        const float inf2 = be * s2 * i2;
        const float k2s = -inf2, k2i = inf2 - GAMMA_C * i2, k2r = GAMMA_C * i2;
        const float s3 = sv + hdt * k2s, i3 = iv + hdt * k2i;
        const float inf3 = be * s3 * i3;
        const float k3s = -inf3, k3i = inf3 - GAMMA_C * i3, k3r = GAMMA_C * i3;
        const float s4 = sv + dt * k3s, i4 = iv + dt * k3i;
        const float inf4 = be * s4 * i4;
        const float k4s = -inf4, k4i = inf4 - GAMMA_C * i4, k4r = GAMMA_C * i4;
        S[rr] = sv    + c6 * (k1s + 2.f * k2s + 2.f * k3s + k4s);
        I[rr] = iv    + c6 * (k1i + 2.f * k2i + 2.f * k3i + k4i);
        R[rr] = R[rr] + c6 * (k1r + 2.f * k2r + 2.f * k3r + k4r);
      }
    }
  }
}

extern "C" void kernel_launch(void* const* d_in, const int* in_sizes, int n_in,
                              void* d_out, int out_size, void* d_ws, size_t ws_size,
                              hipStream_t stream) {
  const float* x   = (const float*)d_in[0];
  const float* t   = (const float*)d_in[1];
  const float* Ws1 = (const float*)d_in[2];
  const float* bs1 = (const float*)d_in[3];
  const float* W3  = (const float*)d_in[4];
  const float* b3  = (const float*)d_in[5];
  const float* Ws2 = (const float*)d_in[6];
  const float* bs2 = (const float*)d_in[7];
  float* out = (float*)d_out;

  // 4000 rows / 16 rows-per-wave = 250 waves; 2 waves per 64-thread block.
  dim3 grid(125), block(64);
  sir_node_fused_kernel<<<grid, block, 0, stream>>>(x, t, Ws1, bs1, W3, b3,
                                                    Ws2, bs2, out);
}